// AlphaFNO2d_74363063763649
// MI455X (gfx1250) — compile-verified
//
#include <hip/hip_runtime.h>
#include <math.h>

// ---------------------------------------------------------------------------
// FNO2d forward for MI455X (gfx1250, wave32).
// Heavy GEMMs use V_WMMA_F32_16X16X4_F32 (f32 in/out).
// Spectral convs are computed as sparse-mode direct DFTs (only 24x12 modes
// survive), ~3 GFLOP total -- far cheaper than full FFTs.
// All WMMA LDS operands are pair-interleaved so each lane fragment is one
// aligned 8-byte unit -> single ds_load_b64 per fragment, no v_mov assembly.
// ---------------------------------------------------------------------------

typedef __attribute__((ext_vector_type(2))) float v2f;
typedef __attribute__((ext_vector_type(8))) float v8f;

#define B_ 8
#define C_ 32
#define H_ 256
#define W_ 256
#define S_ 247
#define M1 12
#define M2 12
#define NMODE (2 * M1 * M2) /* 288: ky in {0..11, 244..255}, kx in {0..11} */
#define TWO_PI_256 0.024543692606170259f /* 2*pi/256 */

// D = A(16x4) * B(4x16) + C, f32 WMMA.
// Lane layouts (ISA 7.12.2):
//  A: lane l<16 holds A[l][k0],A[l][k0+1]; lane l+16 holds A[l][k0+2],A[l][k0+3]
//  B: vgpr0 = B[k0][n] (lanes0-15) / B[k0+2][n] (lanes16-31); vgpr1 = k0+1 / k0+3
//  C/D: vgpr v, lane l<16 -> (M=v,   N=l); lane l+16 -> (M=v+8, N=l)
__device__ __forceinline__ v8f wmma_f32(v2f a, v2f b, v8f c) {
  return __builtin_amdgcn_wmma_f32_16x16x4_f32(false, a, false, b, (short)0, c,
                                               false, false);
}

// Interleaved index for the K-dimension: element (k, n) of an operand with
// row-length N lives at float index ((k>>2)*2 + ((k>>1)&1)) * N * 2 + n*2 + (k&1)
// so that a lane's (k0, k0+1) pair is contiguous 8 bytes.
__device__ __forceinline__ int ilv(int k, int n, int N) {
  return (((k >> 2) * 2 + ((k >> 1) & 1)) * N + n) * 2 + (k & 1);
}

// ---------------------------------------------------------------------------
// DFT basis for W-direction rfft, written directly in WMMA-interleaved layout.
// Logical basis: Dw[k][2j]=cos(2pi k j/256), Dw[k][2j+1]=-sin, j<12; cols>=24: 0
// ---------------------------------------------------------------------------
__global__ void k_init_basis(float* __restrict__ Dw) {
  int k = threadIdx.x; // 0..255
  for (int n = 0; n < 32; ++n) {
    float v = 0.0f;
    if (n < 24) {
      int j = n >> 1;
      float s, c;
      __sincosf((float)((k * j) & 255) * TWO_PI_256, &s, &c);
      v = (n & 1) ? -s : c;
    }
    Dw[ilv(k, n, 32)] = v;
  }
}

// ---------------------------------------------------------------------------
// incremental_x (interp along W) + fc0 (1->32) + transpose to BCHW + zero-pad
// ---------------------------------------------------------------------------
__global__ void k_prep(const float* __restrict__ x, const float* __restrict__ fc0w,
                       const float* __restrict__ fc0b, float* __restrict__ A) {
  int tid = blockIdx.x * 256 + threadIdx.x; // over B*H*W = 524288
  int w = tid & 255, h = (tid >> 8) & 255, b = tid >> 16;
  bool interior = (h < S_) && (w < S_);
  float g = 0.0f;
  if (interior) {
    const float* xb = x + ((size_t)b * S_ + h) * S_;
    if ((w & 1) == 0) g = xb[w >> 1];
    else              g = 0.5f * (xb[(w - 1) >> 1] + xb[(w + 1) >> 1]);
  }
  float* dst = A + (size_t)b * C_ * H_ * W_ + h * W_ + w;
  for (int c = 0; c < C_; ++c)
    dst[(size_t)c * (H_ * W_)] = interior ? (g * fc0w[c] + fc0b[c]) : 0.0f;
}

// ---------------------------------------------------------------------------
// Forward DFT along W: (B*C*H rows, 256) x (256, 24) via WMMA f32 16x16x4.
// One wave per 16-row tile; 128 wmma per wave; fragments are single b64 loads.
// FW layout: [b][c][h][24] (12 complex: re,im interleaved)
// ---------------------------------------------------------------------------
__global__ void __launch_bounds__(32) k_dft_w(const float* __restrict__ act,
                                              const float* __restrict__ Dw,
                                              float* __restrict__ FW) {
  __shared__ float Xs[16 * 256]; // 16 rows of data (A-operand, naturally paired)
  __shared__ float Bs[256 * 32]; // basis, pre-interleaved in global
  int lane = threadIdx.x;
  int ht = blockIdx.x, c = blockIdx.y, b = blockIdx.z;
  const float* row0 = act + ((((size_t)b * C_ + c) * H_) + ht * 16) * W_;
  for (int i = lane; i < (16 * 256) / 4; i += 32)
    ((float4*)Xs)[i] = ((const float4*)row0)[i];
  for (int i = lane; i < (256 * 32) / 4; i += 32)
    ((float4*)Bs)[i] = ((const float4*)Dw)[i];
  __syncthreads();

  int half = lane >> 4, l16 = lane & 15;
  v8f acc0 = {}, acc1 = {};
  for (int k = 0; k < 256; k += 4) {
    int g2 = (k >> 2) * 2 + half; // interleaved group row
    v2f a = *(const v2f*)&Xs[l16 * 256 + k + 2 * half];
    const float* brow = &Bs[g2 * 64];
    v2f b0 = *(const v2f*)&brow[l16 * 2];
    v2f b1 = *(const v2f*)&brow[(16 + l16) * 2];
    acc0 = wmma_f32(a, b0, acc0);
    acc1 = wmma_f32(a, b1, acc1);
  }
  float* out = FW + ((((size_t)b * C_ + c) * H_) + ht * 16) * 24;
  for (int v = 0; v < 8; ++v) {
    int m = v + 8 * half;
    out[m * 24 + l16] = acc0[v];
    if (16 + l16 < 24) out[m * 24 + 16 + l16] = acc1[v];
  }
}

// ---------------------------------------------------------------------------
// DFT along H on 24 kept rows: FH[b][c][m][2], m = kyi*12+kx, kyi<12 -> ky=kyi,
// else ky = 244+(kyi-12). 288 threads/block, one block per (b,c).
// ---------------------------------------------------------------------------
__global__ void k_dft_h(const float* __restrict__ FW, float* __restrict__ FH) {
  __shared__ float Fs[256 * 24];
  int c = blockIdx.x, b = blockIdx.y, t = threadIdx.x;
  const float* src = FW + ((size_t)b * C_ + c) * H_ * 24;
  for (int i = t; i < (256 * 24) / 4; i += 288)
    ((float4*)Fs)[i] = ((const float4*)src)[i];
  __syncthreads();

  int kyi = t / 12, kx = t % 12;
  int ky = (kyi < M1) ? kyi : (232 + kyi); // 244..255
  float re = 0.0f, im = 0.0f;
  for (int h = 0; h < 256; ++h) {
    float s, cth;
    __sincosf((float)((h * ky) & 255) * TWO_PI_256, &s, &cth);
    v2f f = *(const v2f*)&Fs[h * 24 + 2 * kx];
    re += f.x * cth + f.y * s;   // (fr+ifi)*e^{-i th}
    im += f.y * cth - f.x * s;
  }
  float* dst = FH + (((size_t)b * C_ + c) * NMODE + t) * 2;
  dst[0] = re; dst[1] = im;
}

// ---------------------------------------------------------------------------
// Per-mode complex channel mix: O[b][o][m] = sum_i FH[b][i][m] * W[i][o][ky][kx]
// ---------------------------------------------------------------------------
__global__ void k_mode_mix(const float* __restrict__ FH,
                           const float* __restrict__ w1r, const float* __restrict__ w1i,
                           const float* __restrict__ w2r, const float* __restrict__ w2i,
                           float* __restrict__ O) {
  __shared__ float Fs[C_ * 2];
  int m = blockIdx.x, b = blockIdx.y, o = threadIdx.x;
  int kyi = m / 12, kx = m % 12;
  const float *wr, *wi; int ky;
  if (kyi < M1) { wr = w1r; wi = w1i; ky = kyi; }
  else          { wr = w2r; wi = w2i; ky = kyi - M1; }
  const float* src = FH + (((size_t)b * C_ + o) * NMODE + m) * 2; // thread o loads i=o
  Fs[2 * o] = src[0]; Fs[2 * o + 1] = src[1];
  __syncthreads();
  float re = 0.0f, im = 0.0f;
  for (int i = 0; i < C_; ++i) {
    v2f f = *(const v2f*)&Fs[2 * i];
    int idx = ((i * C_ + o) * M1 + ky) * M2 + kx;
    float a = wr[idx], bb = wi[idx];
    re += f.x * a - f.y * bb;
    im += f.x * bb + f.y * a;
  }
  float* dst = O + (((size_t)b * C_ + o) * NMODE + m) * 2;
  dst[0] = re; dst[1] = im;
}

// ---------------------------------------------------------------------------
// Inverse DFT along H: T[b][c][h][kx] = sum_ky O[ky,kx] e^{+2pi i h ky/256}
// ---------------------------------------------------------------------------
__global__ void k_idft_h(const float* __restrict__ O, float* __restrict__ T) {
  __shared__ float Os[NMODE * 2];
  int c = blockIdx.x, b = blockIdx.y, h = threadIdx.x;
  const float* src = O + ((size_t)b * C_ + c) * NMODE * 2;
  for (int i = h; i < NMODE * 2; i += 256) Os[i] = src[i];
  __syncthreads();
  float re[M2] = {}, im[M2] = {};
  for (int kyi = 0; kyi < 2 * M1; ++kyi) {
    int ky = (kyi < M1) ? kyi : (232 + kyi);
    float s, cth;
    __sincosf((float)((h * ky) & 255) * TWO_PI_256, &s, &cth);
    for (int kx = 0; kx < M2; ++kx) {
      v2f o = *(const v2f*)&Os[(kyi * M2 + kx) * 2];
      re[kx] += o.x * cth - o.y * s;
      im[kx] += o.x * s + o.y * cth;
    }
  }
  float* dst = T + (((size_t)b * C_ + c) * H_ + h) * 24;
  for (int kx = 0; kx < M2; ++kx) { dst[2 * kx] = re[kx]; dst[2 * kx + 1] = im[kx]; }
}

// ---------------------------------------------------------------------------
// Fused: inverse-rfft along W (12 cosine terms) + pointwise 32x32 conv (WMMA)
// + optional tanh. One wave per (b, h, 16-pixel tile); 16 wmma per wave.
// B-operand (activation tile) is stored channel-pair-interleaved in LDS.
// ---------------------------------------------------------------------------
template <bool TANH>
__global__ void __launch_bounds__(32) k_spec_pw(const float* __restrict__ actIn,
                                                const float* __restrict__ T,
                                                const float* __restrict__ Wp, // [o][c]
                                                const float* __restrict__ bp,
                                                float* __restrict__ actOut) {
  __shared__ float Xs[C_ * 16];  // interleaved: ilv(c, pix, 16)
  __shared__ float Ws[C_ * C_];  // [o][c] flat (A-operand pairs contiguous in c)
  __shared__ float Ts[C_ * 24];  // [c][2kx]
  int lane = threadIdx.x;
  int wt = blockIdx.x, h = blockIdx.y, b = blockIdx.z;
  int w0 = wt * 16;

  // load this lane's channel row (c = lane), scatter into interleaved layout
  const float* aIn = actIn + (((size_t)b * C_ + lane) * H_ + h) * W_ + w0;
  float xr[16];
  for (int i = 0; i < 16; i += 4) *(float4*)&xr[i] = *(const float4*)&aIn[i];
  {
    int cg = (lane >> 2) * 2 + ((lane >> 1) & 1), comp = lane & 1;
    for (int i = 0; i < 16; ++i) Xs[(cg * 16 + i) * 2 + comp] = xr[i];
  }
  for (int i = lane; i < (C_ * C_) / 4; i += 32)
    ((float4*)Ws)[i] = ((const float4*)Wp)[i];
  const float* tIn = T + (((size_t)b * C_ + lane) * H_ + h) * 24;
  for (int i = 0; i < 24; i += 4)
    *(float4*)&Ts[lane * 24 + i] = *(const float4*)&tIn[i];
  __syncthreads();

  int half = lane >> 4, l16 = lane & 15;
  v8f acc0 = {}, acc1 = {};
  for (int k = 0; k < C_; k += 4) {
    int k0 = k + 2 * half;
    v2f bf = *(const v2f*)&Xs[(((k >> 2) * 2 + half) * 16 + l16) * 2];
    v2f a0 = *(const v2f*)&Ws[l16 * C_ + k0];
    v2f a1 = *(const v2f*)&Ws[(16 + l16) * C_ + k0];
    acc0 = wmma_f32(a0, bf, acc0);
    acc1 = wmma_f32(a1, bf, acc1);
  }

  // irfft-W cosine factors for this lane's pixel (period 256)
  int w = w0 + l16;
  float cw[M2], sw[M2];
  for (int kx = 0; kx < M2; ++kx)
    __sincosf((float)((w * kx) & 255) * TWO_PI_256, &sw[kx], &cw[kx]);

  const float inv = 1.0f / 65536.0f; // 1/(H*W) from ifft+irfft
  for (int mt = 0; mt < 2; ++mt) {
    v8f acc = mt ? acc1 : acc0;
    for (int v = 0; v < 8; ++v) {
      int ch = mt * 16 + v + 8 * half;
      float s = 0.0f;
      for (int kx = 0; kx < M2; ++kx) {
        v2f tt = *(const v2f*)&Ts[ch * 24 + 2 * kx];
        float term = tt.x * cw[kx] - tt.y * sw[kx];
        s += (kx ? 2.0f : 1.0f) * term;
      }
      float val = acc[v] + bp[ch] + s * inv;
      if (TANH) val = tanhf(val);
      actOut[(((size_t)b * C_ + ch) * H_ + h) * W_ + w] = val;
    }
  }
}

// ---------------------------------------------------------------------------
// Per-channel BN training statistics over (B,H,W): stats[c]=mean, stats[32+c]=var
// ---------------------------------------------------------------------------
__global__ void k_bnstats(const float* __restrict__ Y, float* __restrict__ stats) {
  __shared__ float rs[256], rs2[256];
  int c = blockIdx.x, t = threadIdx.x;
  float s = 0.0f, s2 = 0.0f;
  for (int b = 0; b < B_; ++b) {
    const float* p = Y + ((size_t)b * C_ + c) * (H_ * W_);
    for (int i = t; i < H_ * W_; i += 256) { float v = p[i]; s += v; s2 += v * v; }
  }
  rs[t] = s; rs2[t] = s2;
  __syncthreads();
  for (int o = 128; o > 0; o >>= 1) {
    if (t < o) { rs[t] += rs[t + o]; rs2[t] += rs2[t + o]; }
    __syncthreads();
  }
  if (t == 0) {
    const float n = (float)(B_ * H_ * W_);
    float mean = rs[0] / n;
    stats[c] = mean;
    stats[C_ + c] = rs2[0] / n - mean * mean;
  }
}

// ---------------------------------------------------------------------------
// Fused head: BN+tanh -> fc1(32->128, WMMA) + tanh -> fc2(128->1) -> alpha-sigmoid
// Block = 8 waves; each wave one 16-pixel tile; 64 wmma per wave.
// Both fc1 weights (A) and activation tile (B) are pair-interleaved in LDS.
// ---------------------------------------------------------------------------
__global__ void __launch_bounds__(256) k_head(
    const float* __restrict__ Y, const float* __restrict__ stats,
    const float* __restrict__ bng, const float* __restrict__ bnb,
    const float* __restrict__ fc1w /*[32][128]*/, const float* __restrict__ fc1b,
    const float* __restrict__ fc2w, const float* __restrict__ fc2b,
    float* __restrict__ out) {
  __shared__ float W1[C_ * 128];   // interleaved: ilv(c, hid, 128)
  __shared__ float Xs[8][C_ * 16]; // interleaved: ilv(c, pix, 16) per wave
  __shared__ float sc[C_], sh[C_];
  int t = threadIdx.x, lane = t & 31, wave = t >> 5;
  int b = blockIdx.z, h = blockIdx.y;
  int w0 = (blockIdx.x * 8 + wave) * 16;

  for (int i = t; i < C_ * 128; i += 256) {
    int c = i >> 7, hid = i & 127;
    W1[ilv(c, hid, 128)] = fc1w[i];
  }
  if (t < C_) {
    float inv = rsqrtf(stats[C_ + t] + 1e-5f);
    sc[t] = bng[t] * inv;
    sh[t] = bnb[t] - stats[t] * inv * bng[t];
  }
  float yreg[16];
  const float* yp = Y + (((size_t)b * C_ + lane) * H_ + h) * W_ + w0;
  for (int i = 0; i < 16; ++i) yreg[i] = yp[i];
  __syncthreads();
  {
    int cg = (lane >> 2) * 2 + ((lane >> 1) & 1), comp = lane & 1;
    for (int i = 0; i < 16; ++i)
      Xs[wave][(cg * 16 + i) * 2 + comp] = tanhf(yreg[i] * sc[lane] + sh[lane]);
  }
  __syncthreads();

  int half = lane >> 4, l16 = lane & 15;
  v8f acc[8];
  { v8f z = {}; for (int i = 0; i < 8; ++i) acc[i] = z; }
  for (int k = 0; k < C_; k += 4) {
    int g2 = (k >> 2) * 2 + half;
    v2f bf = *(const v2f*)&Xs[wave][(g2 * 16 + l16) * 2];
    for (int nt = 0; nt < 8; ++nt) {
      v2f a = *(const v2f*)&W1[(g2 * 128 + nt * 16 + l16) * 2];
      acc[nt] = wmma_f32(a, bf, acc[nt]);
    }
  }
  float p = 0.0f;
  for (int nt = 0; nt < 8; ++nt)
    for (int v = 0; v < 8; ++v) {
      int hid = nt * 16 + v + 8 * half;
      p += tanhf(acc[nt][v] + fc1b[hid]) * fc2w[hid];
    }
  p += __shfl_xor(p, 16); // combine hidden halves held by lane l and l+16
  if (half == 0) {
    int w = w0 + l16;
    if (w < S_) {
      float z = p + fc2b[0];
      float sig = 1.0f / (1.0f + expf(-z));
      out[((size_t)b * S_ + h) * S_ + w] = 0.05f + 0.95f * sig;
    }
  }
}

// ---------------------------------------------------------------------------
// Launch
// ---------------------------------------------------------------------------
extern "C" void kernel_launch(void* const* d_in, const int* in_sizes, int n_in,
                              void* d_out, int out_size, void* d_ws, size_t ws_size,
                              hipStream_t stream) {
  (void)in_sizes; (void)n_in; (void)out_size; (void)ws_size;
  const float* x     = (const float*)d_in[0];
  const float* fc0w  = (const float*)d_in[1];
  const float* fc0b  = (const float*)d_in[2];
  const float* c0w1r = (const float*)d_in[3];
  const float* c0w1i = (const float*)d_in[4];
  const float* c0w2r = (const float*)d_in[5];
  const float* c0w2i = (const float*)d_in[6];
  const float* c1w1r = (const float*)d_in[7];
  const float* c1w1i = (const float*)d_in[8];
  const float* c1w2r = (const float*)d_in[9];
  const float* c1w2i = (const float*)d_in[10];
  const float* w0w   = (const float*)d_in[11];
  const float* w0b   = (const float*)d_in[12];
  const float* w1w   = (const float*)d_in[13];
  const float* w1b   = (const float*)d_in[14];
  const float* bng   = (const float*)d_in[15];
  const float* bnb   = (const float*)d_in[16];
  const float* fc1w  = (const float*)d_in[17];
  const float* fc1b  = (const float*)d_in[18];
  const float* fc2w  = (const float*)d_in[19];
  const float* fc2b  = (const float*)d_in[20];
  float* out = (float*)d_out;

  // Scratch layout (~142 MB total; every byte read is written first).
  char* ws = (char*)d_ws;
  const size_t SZ_ACT = (size_t)B_ * C_ * H_ * W_ * 4;   // 67108864
  const size_t SZ_FW  = (size_t)B_ * C_ * H_ * 24 * 4;   // 6291456
  const size_t SZ_FH  = (size_t)B_ * C_ * NMODE * 2 * 4; // 589824
  float* bufA  = (float*)(ws);                          // A0, later pre-BN Y
  float* bufB  = (float*)(ws + SZ_ACT);                 // layer0 output
  float* bufFW = (float*)(ws + 2 * SZ_ACT);             // FW, reused as T
  float* bufFH = (float*)(ws + 2 * SZ_ACT + SZ_FW);
  float* bufO  = (float*)(ws + 2 * SZ_ACT + SZ_FW + SZ_FH);
  float* bufDw = (float*)(ws + 2 * SZ_ACT + SZ_FW + 2 * SZ_FH);
  float* stats = (float*)(ws + 2 * SZ_ACT + SZ_FW + 2 * SZ_FH + 256 * 32 * 4);

  k_init_basis<<<1, 256, 0, stream>>>(bufDw);
  k_prep<<<(B_ * H_ * W_) / 256, 256, 0, stream>>>(x, fc0w, fc0b, bufA);

  // ---- Layer 0 ----
  k_dft_w<<<dim3(16, C_, B_), 32, 0, stream>>>(bufA, bufDw, bufFW);
  k_dft_h<<<dim3(C_, B_), 288, 0, stream>>>(bufFW, bufFH);
  k_mode_mix<<<dim3(NMODE, B_), C_, 0, stream>>>(bufFH, c0w1r, c0w1i, c0w2r, c0w2i, bufO);
  k_idft_h<<<dim3(C_, B_), 256, 0, stream>>>(bufO, bufFW); // T overwrites FW
  k_spec_pw<true><<<dim3(16, H_, B_), 32, 0, stream>>>(bufA, bufFW, w0w, w0b, bufB);

  // ---- Layer 1 ----
  k_dft_w<<<dim3(16, C_, B_), 32, 0, stream>>>(bufB, bufDw, bufFW);
  k_dft_h<<<dim3(C_, B_), 288, 0, stream>>>(bufFW, bufFH);
  k_mode_mix<<<dim3(NMODE, B_), C_, 0, stream>>>(bufFH, c1w1r, c1w1i, c1w2r, c1w2i, bufO);
  k_idft_h<<<dim3(C_, B_), 256, 0, stream>>>(bufO, bufFW);
  k_spec_pw<false><<<dim3(16, H_, B_), 32, 0, stream>>>(bufB, bufFW, w1w, w1b, bufA);

  // ---- BN stats + fused head ----
  k_bnstats<<<C_, 256, 0, stream>>>(bufA, stats);
  k_head<<<dim3(2, S_, B_), 256, 0, stream>>>(bufA, stats, bng, bnb,
                                              fc1w, fc1b, fc2w, fc2b, out);
}